// PhysicsResonatorBank_83004537962849
// MI455X (gfx1250) — compile-verified
//
#include <hip/hip_runtime.h>
#include <hip/hip_bf16.h>
#include <math.h>

#define KPART 80
#define NB    16
#define BATCH 8
#define TWO_PI 6.28318530717958647692f

typedef __attribute__((ext_vector_type(2))) float v2f;
typedef __attribute__((ext_vector_type(8))) float v8f;

// ---------------------------------------------------------------------------
// ws layout (floats) — per-k params first and contiguous so the synth kernel
// can stage them into LDS with one coalesced loop:
//   [0)      gL   [K]
//   [K)      gR   [K]        (contiguous with gL -> branchless lane select)
//   [2K)     t1i  [K]        1/tau1
//   [3K)     t2i  [K]        1/tau2
//   [4K)     a1s  [K]        sigmoid(logit_a1)
//   [5K)     bhz  [K]        beat Hz
//   [6K)     dep  [K]        0.5*sigmoid(logit_beat_depth)
//   [7K)     fhz  [B*K]      partial freq in Hz
//   [7K+BK)  ampv [B*K]      exp(log_A0)*vel*alive
//   [7K+2BK) nfhz [NB], bg [NB], nsc [B], tauNi [1]
// ---------------------------------------------------------------------------
#define WS_FHZ   (7 * KPART)
#define WS_AMP   (7 * KPART + BATCH * KPART)
#define WS_NOISE (7 * KPART + 2 * BATCH * KPART)

__global__ void prb_setup_kernel(
    const float* __restrict__ f0, const float* __restrict__ vel,
    const float* __restrict__ slope, const float* __restrict__ inter,
    const float* __restrict__ ltau1, const float* __restrict__ ltau2,
    const float* __restrict__ la1, const float* __restrict__ lbhz,
    const float* __restrict__ lbd, const float* __restrict__ lA0,
    const float* __restrict__ det, const float* __restrict__ nbg,
    const float* __restrict__ ltauN, const float* __restrict__ lnl,
    const float* __restrict__ pansc, float* __restrict__ ws)
{
    float* gL   = ws;
    float* gR   = ws + KPART;
    float* t1i  = ws + 2 * KPART;
    float* t2i  = ws + 3 * KPART;
    float* a1s  = ws + 4 * KPART;
    float* bhzA = ws + 5 * KPART;
    float* depA = ws + 6 * KPART;
    float* fhz  = ws + WS_FHZ;
    float* ampv = ws + WS_AMP;
    float* nfhz = ws + WS_NOISE;
    float* bg   = nfhz + NB;
    float* nsc  = bg + NB;
    float* tauNi = nsc + BATCH;

    const int tid = threadIdx.x;

    for (int idx = tid; idx < BATCH * KPART; idx += blockDim.x) {
        int b = idx / KPART, k = idx - b * KPART;
        float fb   = f0[b];
        float midi = 69.0f + 12.0f * __log2f(fb * (1.0f / 440.0f));
        float mn   = (midi - 21.0f) * (1.0f / 87.0f);
        float lb   = fminf(fmaxf(slope[0] * mn + inter[0], -14.0f), -4.0f);
        float Bi   = __expf(lb);
        float kk   = (float)(k + 1);
        float f    = fb * kk * sqrtf(1.0f + Bi * kk * kk) * (1.0f + det[k]);
        float alive = (f < 22050.0f) ? 1.0f : 0.0f;
        fhz[idx]  = f;
        ampv[idx] = __expf(lA0[k]) * vel[b] * alive;
    }

    for (int k = tid; k < KPART; k += blockDim.x) {
        t1i[k]  = __expf(-ltau1[k]);
        t2i[k]  = __expf(-ltau2[k]);
        a1s[k]  = 1.0f / (1.0f + __expf(-la1[k]));
        bhzA[k] = __expf(lbhz[k]);
        depA[k] = 0.5f / (1.0f + __expf(-lbd[k]));
        float pinit = ((float)k / (float)(KPART - 1) * 2.0f - 1.0f) * 0.1f;
        float pan   = tanhf(pinit * __expf(pansc[0]));
        gL[k] = sqrtf(0.5f * (1.0f - pan));
        gR[k] = sqrtf(0.5f * (1.0f + pan));
    }

    if (tid == 0) {
        const float l0 = 5.29831736654804f;   // log(200)
        const float l1 = 9.90348755253613f;   // log(20000)
        float mx = -1e30f;
        for (int i = 0; i < NB; ++i) mx = fmaxf(mx, nbg[i]);
        float e[NB]; float ssum = 0.0f;
        for (int i = 0; i < NB; ++i) { e[i] = __expf(nbg[i] - mx); ssum += e[i]; }
        float inv = 1.0f / ssum;
        for (int i = 0; i < NB; ++i) {
            float f = __expf(l0 + (l1 - l0) * ((float)i / (float)(NB - 1)));
            nfhz[i] = f;
            bg[i]   = e[i] * inv * ((f < 22050.0f) ? 1.0f : 0.0f);
        }
        tauNi[0] = __expf(-ltauN[0]);
        float nl = __expf(lnl[0]);
        for (int b = 0; b < BATCH; ++b) nsc[b] = nl * vel[b];
    }
}

// ---------------------------------------------------------------------------
// Main synth kernel. One wave32 = one (batch, 16-sample) tile.
// All per-k tables staged in LDS once per block (all 8 waves share one b).
// Fully unrolled K loop: 20x v_wmma_f32_16x16x4_f32 chained on the f32
// accumulator; A operand = per-lane partial evaluations (trans-heavy),
// B operand = pan gains via branchless LDS select (no EXEC divergence).
// ---------------------------------------------------------------------------
__global__ void prb_synth_kernel(const float* __restrict__ ws,
                                 float* __restrict__ out,
                                 int N, int tilesPerB)
{
    __shared__ float sPar[7 * KPART];   // gL,gR,t1i,t2i,a1s,bhz,dep
    __shared__ float sFA[2 * KPART];    // fhz(b,:), ampv(b,:)
    __shared__ float shn[8][16];        // per-wave per-sample noise

    const int tid = threadIdx.x;
    const int wavesPerBlock = blockDim.x >> 5;
    const int b = (blockIdx.x * wavesPerBlock) / tilesPerB;  // uniform in block

    for (int i = tid; i < 7 * KPART; i += blockDim.x) sPar[i] = ws[i];
    for (int i = tid; i < KPART; i += blockDim.x) {
        sFA[i]         = ws[WS_FHZ + b * KPART + i];
        sFA[KPART + i] = ws[WS_AMP + b * KPART + i];
    }
    __syncthreads();

    const float* sT1 = sPar + 2 * KPART;
    const float* sT2 = sPar + 3 * KPART;
    const float* sA1 = sPar + 4 * KPART;
    const float* sBH = sPar + 5 * KPART;
    const float* sDP = sPar + 6 * KPART;
    const float* sFz = sFA;
    const float* sAm = sFA + KPART;

    const int lane = tid & 31;
    const int wv   = tid >> 5;
    const int w    = blockIdx.x * wavesPerBlock + wv;
    const int tile = w - b * tilesPerB;
    const int n0   = tile * 16;
    const int h    = lane >> 4;   // selects k-pair within the 4-wide K step
    const int m    = lane & 15;   // sample row (A) / output column (B,D)

    const float t = (float)(n0 + m) * (1.0f / 44100.0f);

    // Branchless gain fragment: gR = gL + K in LDS; lanes m>=2 read gL but
    // multiply by 0. Unconditional LDS reads -> EXEC stays all-ones.
    const float  gmask = (m < 2) ? 1.0f : 0.0f;
    const float* gsel  = sPar + ((m == 1) ? KPART : 0);

    v8f acc = {};
    #pragma unroll
    for (int kc = 0; kc < KPART; kc += 4) {
        const int k0 = kc + 2 * h;
        v2f af, bf;
        #pragma unroll
        for (int j = 0; j < 2; ++j) {
            const int k = k0 + j;
            float a1 = sA1[k];
            float env = a1 * __expf(-t * sT1[k]) +
                        (1.0f - a1) * __expf(-t * sT2[k]);
            float beat = 1.0f + sDP[k] * (__cosf(TWO_PI * sBH[k] * t) - 1.0f);
            float cyc = sFz[k] * t;          // phase in cycles
            cyc -= floorf(cyc);              // range-reduce before sin
            float p = sAm[k] * env * beat * __sinf(TWO_PI * cyc);
            if (j == 0) af.x = p; else af.y = p;
        }
        bf.x = gmask * gsel[k0];
        bf.y = gmask * gsel[k0 + 1];
        acc = __builtin_amdgcn_wmma_f32_16x16x4_f32(
                  false, af, false, bf, (short)0, acc, false, false);
    }

    // Attack noise for this sample (channel-independent), shared via LDS so
    // the channel-writer lanes can fetch it per output row.
    const float* nfhz  = ws + WS_NOISE;
    const float* bg    = nfhz + NB;
    const float* nsc   = bg + NB;
    const float* tauNi = nsc + BATCH;

    float ns = 0.0f;
    #pragma unroll
    for (int i = 0; i < NB; ++i) {
        float cyc = nfhz[i] * t;
        cyc -= floorf(cyc);
        ns = fmaf(bg[i], __sinf(TWO_PI * cyc), ns);
    }
    ns *= __expf(-t * tauNi[0]);
    float nv = nsc[b] * ns;
    if (h == 0) shn[wv][m] = nv;
    __syncthreads();

    // D layout: lane (N=col=m, half h), VGPR r -> row M = r + 8*h.
    if (m < 2) {
        const size_t base = ((size_t)(b * 2 + m)) * (size_t)N
                          + (size_t)(n0 + 8 * h);
        #pragma unroll
        for (int r = 0; r < 8; ++r)
            out[base + r] = acc[r] + shn[wv][8 * h + r];
    }
}

extern "C" void kernel_launch(void* const* d_in, const int* in_sizes, int n_in,
                              void* d_out, int out_size, void* d_ws, size_t ws_size,
                              hipStream_t stream) {
    const float* f0    = (const float*)d_in[0];
    const float* vel   = (const float*)d_in[1];
    const float* slope = (const float*)d_in[2];
    const float* inter = (const float*)d_in[3];
    const float* ltau1 = (const float*)d_in[4];
    const float* ltau2 = (const float*)d_in[5];
    const float* la1   = (const float*)d_in[6];
    const float* lbhz  = (const float*)d_in[7];
    const float* lbd   = (const float*)d_in[8];
    const float* lA0   = (const float*)d_in[9];
    const float* det   = (const float*)d_in[10];
    const float* nbg   = (const float*)d_in[11];
    const float* ltauN = (const float*)d_in[12];
    const float* lnl   = (const float*)d_in[13];
    const float* pansc = (const float*)d_in[14];
    (void)n_in; (void)in_sizes; (void)ws_size;

    float* ws  = (float*)d_ws;
    float* out = (float*)d_out;

    // out_size = B * 2 * N  ->  N = out_size / 16
    const int N = out_size / (BATCH * 2);
    const int tilesPerB = N / 16;          // multiple of 16 (N = n_frames*256)
    const int totalWaves = BATCH * tilesPerB;
    const int blocks = totalWaves / 8;     // 8 waves (256 threads) per block

    prb_setup_kernel<<<1, 256, 0, stream>>>(f0, vel, slope, inter, ltau1, ltau2,
                                            la1, lbhz, lbd, lA0, det, nbg,
                                            ltauN, lnl, pansc, ws);
    prb_synth_kernel<<<blocks, 256, 0, stream>>>(ws, out, N, tilesPerB);
}